// SparseArch_61057255079950
// MI455X (gfx1250) — compile-verified
//
#include <hip/hip_runtime.h>
#include <hip/hip_bf16.h>

typedef __attribute__((ext_vector_type(2))) float v2f;
typedef __attribute__((ext_vector_type(8))) float v8f;

#define BATCH   8192
#define LFACT   40
#define ZCH     2000
#define DIM     128
#define NITEMS  (BATCH * LFACT)            // 327680 per table
#define SCALE   (1.0f / (float)(BATCH * 2 * DIM))  // 1 / 2097152

// ---------------------------------------------------------------------------
// Kernel 0: zero the scalar output (d_out is poisoned by the harness).
// ---------------------------------------------------------------------------
__global__ void zero_out_kernel(float* out) { out[0] = 0.0f; }

// ---------------------------------------------------------------------------
// Kernel 1: rowsum via V_WMMA_F32_16X16X4_F32.
//   rowsum[z] = sum_d weight[z,d]  ==  (W_tile 16x4) x (ones 4x16), accumulated
//   over D in 32 steps. One wave (32 threads) per 16-row tile.
//   A-matrix f32 16x4 layout: lanes 0-15 -> M=lane, K=0..1 in VGPR0..1;
//                             lanes 16-31 -> M=lane-16, K=2..3.
//   C/D layout: VGPR v: lanes 0-15 hold (M=v, N=lane), lanes 16-31 (M=v+8).
//   With B = ones, every column of D equals the row sums, so lane 0 carries
//   rowsum[M=0..7] and lane 16 carries rowsum[M=8..15] in acc[0..7].
// ---------------------------------------------------------------------------
__global__ __launch_bounds__(32) void rowsum_wmma_kernel(
    const float* __restrict__ w0, const float* __restrict__ w1,
    float* __restrict__ rs /* [2*ZCH] */) {
  const int blk   = blockIdx.x;              // 0..249
  const int table = (blk >= (ZCH / 16));     // 125 tiles per table
  const float* w  = table ? w1 : w0;
  float* out      = rs + (table ? ZCH : 0);
  const int rowbase = (blk - table * (ZCH / 16)) * 16;

  const int lane = threadIdx.x & 31;
  const int m    = lane & 15;
  const int kb   = (lane >> 4) << 1;         // 0 or 2
  const float* rowp = w + (size_t)(rowbase + m) * DIM;

  v8f acc = {};
  v2f ones; ones.x = 1.0f; ones.y = 1.0f;

#pragma unroll
  for (int d = 0; d < DIM; d += 4) {
    v2f a;
    a.x = rowp[d + kb];
    a.y = rowp[d + kb + 1];
    // D = A(16x4) * ones(4x16) + C   -> v_wmma_f32_16x16x4_f32
    acc = __builtin_amdgcn_wmma_f32_16x16x4_f32(
        /*neg_a=*/false, a, /*neg_b=*/false, ones,
        /*c_mod=*/(short)0, acc, /*reuse_a=*/false, /*reuse_b=*/false);
  }

  if (lane == 0) {
#pragma unroll
    for (int v = 0; v < 8; ++v) out[rowbase + v] = acc[v];
  } else if (lane == 16) {
#pragma unroll
    for (int v = 0; v < 8; ++v) out[rowbase + 8 + v] = acc[v];
  }
}

// ---------------------------------------------------------------------------
// Kernel 2: binary-search remap + pooled reduction to one scalar.
//   LDS-cache both identity tables (sorted int32) and both rowsum vectors
//   (32,000 B total; WGP has 320 KB). Each i in [0, B*L) contributes one
//   lookup per table. Wave32 shuffle reduce -> LDS reduce -> one atomicAdd.
// ---------------------------------------------------------------------------
__global__ __launch_bounds__(256) void pool_reduce_kernel(
    const int* __restrict__ ids0, const int* __restrict__ ids1,
    const int* __restrict__ idt0, const int* __restrict__ idt1,
    const float* __restrict__ rs, float* __restrict__ out) {
  __shared__ int   s_id0[ZCH];
  __shared__ int   s_id1[ZCH];
  __shared__ float s_rs0[ZCH];
  __shared__ float s_rs1[ZCH];
  __shared__ float s_wsum[8];                // 256 threads = 8 waves

  for (int i = threadIdx.x; i < ZCH; i += blockDim.x) {
    s_id0[i] = idt0[i];
    s_id1[i] = idt1[i];
    s_rs0[i] = rs[i];
    s_rs1[i] = rs[ZCH + i];
  }
  __syncthreads();

  float acc = 0.0f;
  const int stride = blockDim.x * gridDim.x;
  for (int i = blockIdx.x * blockDim.x + threadIdx.x; i < NITEMS; i += stride) {
    {
      const int id = ids0[i];
      int lo = 0, hi = ZCH;                  // searchsorted(side='left')
      while (lo < hi) {
        const int mid = (lo + hi) >> 1;
        if (s_id0[mid] < id) lo = mid + 1; else hi = mid;
      }
      const int slot = (lo < ZCH - 1) ? lo : (ZCH - 1);
      acc += (s_id0[slot] == id) ? s_rs0[slot] : s_rs0[ZCH - 1];
    }
    {
      const int id = ids1[i];
      int lo = 0, hi = ZCH;
      while (lo < hi) {
        const int mid = (lo + hi) >> 1;
        if (s_id1[mid] < id) lo = mid + 1; else hi = mid;
      }
      const int slot = (lo < ZCH - 1) ? lo : (ZCH - 1);
      acc += (s_id1[slot] == id) ? s_rs1[slot] : s_rs1[ZCH - 1];
    }
  }

  // wave32 reduction
#pragma unroll
  for (int off = 16; off > 0; off >>= 1)
    acc += __shfl_down(acc, off, 32);

  const int lane = threadIdx.x & 31;
  const int wave = threadIdx.x >> 5;
  if (lane == 0) s_wsum[wave] = acc;
  __syncthreads();

  if (threadIdx.x == 0) {
    float bsum = 0.0f;
#pragma unroll
    for (int wv = 0; wv < 8; ++wv) bsum += s_wsum[wv];
    atomicAdd(out, bsum * SCALE);
  }
}

// ---------------------------------------------------------------------------
extern "C" void kernel_launch(void* const* d_in, const int* in_sizes, int n_in,
                              void* d_out, int out_size, void* d_ws, size_t ws_size,
                              hipStream_t stream) {
  const int*   ids0 = (const int*)d_in[0];   // [B, L] int32
  const int*   ids1 = (const int*)d_in[1];   // [B, L] int32
  const int*   idt0 = (const int*)d_in[2];   // [ZCH]  int32, sorted
  const int*   idt1 = (const int*)d_in[3];   // [ZCH]  int32, sorted
  const float* w0   = (const float*)d_in[4]; // [ZCH, D] f32
  const float* w1   = (const float*)d_in[5]; // [ZCH, D] f32
  float* out = (float*)d_out;                // scalar loss
  float* rs  = (float*)d_ws;                 // [2*ZCH] f32 rowsums (16,000 B)

  zero_out_kernel<<<1, 1, 0, stream>>>(out);
  rowsum_wmma_kernel<<<2 * (ZCH / 16), 32, 0, stream>>>(w0, w1, rs);
  pool_reduce_kernel<<<512, 256, 0, stream>>>(ids0, ids1, idt0, idt1, rs, out);
}